// Neuralbeta_27917287424432
// MI455X (gfx1250) — compile-verified
//
#include <hip/hip_runtime.h>
#include <math.h>

#define EPSF 1e-8f

static constexpr int B = 32;
static constexpr int N = 2048;
static constexpr int D = 128;
static constexpr int SPLIT = 16;            // split-K chunks per batch for select reductions
static constexpr int ROWS_PER_CHUNK = N / SPLIT; // 128

typedef __attribute__((ext_vector_type(2))) float v2f;
typedef __attribute__((ext_vector_type(8))) float v8f;

// ---------------- helpers ----------------

__device__ __forceinline__ float wave_sum32(float v) {
  #pragma unroll
  for (int off = 16; off > 0; off >>= 1) v += __shfl_xor(v, off, 32);
  return v;
}

__device__ __forceinline__ float clampA(float a) { return (a > EPSF) ? a : 0.0f; }

struct f4 { float x, y, z, w; };
__device__ __forceinline__ f4 ldg4(const float* p) {
  const float4 v = *reinterpret_cast<const float4*>(p);
  return f4{v.x, v.y, v.z, v.w};
}
__device__ __forceinline__ void stg4(float* p, f4 v) {
  *reinterpret_cast<float4*>(p) = make_float4(v.x, v.y, v.z, v.w);
}
__device__ __forceinline__ float dot4(f4 a, f4 b) {
  return fmaf(a.x, b.x, fmaf(a.y, b.y, fmaf(a.z, b.z, a.w * b.w)));
}
__device__ __forceinline__ f4 lerp4(f4 v, f4 x, float a) {
  // (1-a)*v + a*x
  f4 r;
  r.x = fmaf(a, x.x - v.x, v.x);
  r.y = fmaf(a, x.y - v.y, v.y);
  r.z = fmaf(a, x.z - v.z, v.z);
  r.w = fmaf(a, x.w - v.w, v.w);
  return r;
}

// ---------------- K0: vector norms (at_addr) ----------------

__global__ void __launch_bounds__(128) vec_norm_k(const float* __restrict__ vecs,
                                                  float* __restrict__ norms) {
  const int b = blockIdx.x;
  const int d = threadIdx.x;
  __shared__ float red[128];
  const float v = vecs[b * D + d];
  red[d] = v * v;
  __syncthreads();
  for (int off = 64; off > 0; off >>= 1) {
    if (d < off) red[d] += red[d + off];
    __syncthreads();
  }
  if (d == 0) norms[b] = sqrtf(red[0]);
}

// ---------------- K1: per-row address norms + cs0 ----------------
// wave-per-row; lane handles 4 consecutive d (float4, coalesced)

__global__ void __launch_bounds__(256) rownorm_cs0_k(const float* __restrict__ addrs,
                                                     const float* __restrict__ at_addr,
                                                     const float* __restrict__ at_norm,
                                                     float* __restrict__ addr_norm,
                                                     float* __restrict__ cs0) {
  const int lane = threadIdx.x & 31;
  const int wave = threadIdx.x >> 5;
  const int row = blockIdx.x * 8 + wave;            // < B*N
  const int b = row >> 11;                          // row / N
  const f4 at4 = ldg4(at_addr + (size_t)b * D + lane * 4);
  const f4 v = ldg4(addrs + (size_t)row * D + lane * 4);
  float ss = wave_sum32(dot4(v, v));
  float da = wave_sum32(dot4(v, at4));
  if (lane == 0) {
    const float nrm = sqrtf(ss);
    addr_norm[row] = nrm;
    cs0[row] = da / fmaxf(at_norm[b] * nrm, EPSF);
  }
}

// ---------------- WMMA dots: cs for 2 query vectors ----------------
// D(16x16) tile = A(16 rows x K) * B(K x 16 cols); columns 0/1 = q0/q1, rest 0.
// Emits 32x v_wmma_f32_16x16x4_f32 per 16-row tile.

__global__ void __launch_bounds__(32) wmma_dots2_k(const float* __restrict__ addrs,
                                                   const float* __restrict__ addr_norm,
                                                   const float* __restrict__ q0,
                                                   const float* __restrict__ q1,
                                                   const float* __restrict__ q0n,
                                                   const float* __restrict__ q1n,
                                                   float* __restrict__ csa,
                                                   float* __restrict__ csb) {
  const int lane = threadIdx.x;                 // 0..31, one wave
  const int blk = blockIdx.x;                   // B * (N/16)
  const int b = blk >> 7;                       // / 128
  const int n0 = (blk & 127) << 4;              // tile * 16

  __shared__ __align__(16) float at[16 * 132];  // 16 rows, pitch 132 (bank-conflict free)
  __shared__ __align__(16) float qs[3 * 128];   // q0 | q1 | zeros

  const float* rowbase = addrs + ((size_t)b * N + n0) * D;
  #pragma unroll 4
  for (int r = 0; r < 16; ++r) {
    const float4 v = *reinterpret_cast<const float4*>(rowbase + r * D + lane * 4);
    *reinterpret_cast<float4*>(&at[r * 132 + lane * 4]) = v;
  }
  *reinterpret_cast<float4*>(&qs[lane * 4]) =
      *reinterpret_cast<const float4*>(q0 + (size_t)b * D + lane * 4);
  *reinterpret_cast<float4*>(&qs[128 + lane * 4]) =
      *reinterpret_cast<const float4*>(q1 + (size_t)b * D + lane * 4);
  *reinterpret_cast<float4*>(&qs[256 + lane * 4]) = make_float4(0.f, 0.f, 0.f, 0.f);
  __syncthreads();

  const int col = lane & 15;                    // B/D column for this lane group
  const int mrow = lane & 15;                   // A row (M) for this lane group
  const int koff = (lane < 16) ? 0 : 2;         // K sub-offset per lane half
  const int bcol = (col < 2) ? col : 2;         // columns >=2 read zeros

  v8f c = {0.f, 0.f, 0.f, 0.f, 0.f, 0.f, 0.f, 0.f};
  #pragma unroll 4
  for (int k = 0; k < D; k += 4) {
    v2f a, bb;
    a.x = at[mrow * 132 + k + koff];
    a.y = at[mrow * 132 + k + koff + 1];
    bb.x = qs[bcol * 128 + k + koff];
    bb.y = qs[bcol * 128 + k + koff + 1];
    c = __builtin_amdgcn_wmma_f32_16x16x4_f32(false, a, false, bb, (short)0, c,
                                              false, false);
  }

  // D[m][col]: lanes {0,1} hold rows 0..7 of cols {0,1}; lanes {16,17} rows 8..15.
  if (col < 2) {
    const float qn = (col == 0) ? q0n[b] : q1n[b];
    float* out = (col == 0) ? csa : csb;
    const int m0 = (lane < 16) ? 0 : 8;
    #pragma unroll
    for (int i = 0; i < 8; ++i) {
      const size_t row = (size_t)b * N + n0 + m0 + i;
      out[row] = c[i] / fmaxf(qn * addr_norm[row], EPSF);
    }
  }
}

// ---------------- select_address: split-K partial sums ----------------
// block = (b, split chunk); thread owns one d. Deterministic (no atomics).

__global__ void __launch_bounds__(128) select_split_k(const float* __restrict__ cs,
                                                      const float* __restrict__ v0,
                                                      const float* __restrict__ v1,
                                                      const float* __restrict__ v2,
                                                      float* __restrict__ p0,
                                                      float* __restrict__ p1,
                                                      float* __restrict__ p2) {
  const int b = blockIdx.x / SPLIT;
  const int s = blockIdx.x % SPLIT;
  const int d = threadIdx.x;
  const int r0 = s * ROWS_PER_CHUNK;
  __shared__ float csl[ROWS_PER_CHUNK];
  csl[d] = cs[(size_t)b * N + r0 + d];
  __syncthreads();
  float a0 = 0.f, a1 = 0.f, a2 = 0.f;
  for (int r = 0; r < ROWS_PER_CHUNK; ++r) {
    const float cc = csl[r];
    const size_t off = ((size_t)b * N + r0 + r) * D + d;
    a0 = fmaf(cc, v0[off], a0);
    if (v1) a1 = fmaf(cc, v1[off], a1);
    if (v2) a2 = fmaf(cc, v2[off], a2);
  }
  const size_t po = ((size_t)(b * SPLIT + s)) * D + d;
  p0[po] = a0;
  if (p1) p1[po] = a1;
  if (p2) p2[po] = a2;
}

// reduce partials in fixed order; also emit the result vector's L2 norm.
// grid = ntens * B blocks; t-major.

__global__ void __launch_bounds__(128) select_reduce_k(const float* __restrict__ p0,
                                                       const float* __restrict__ p1,
                                                       const float* __restrict__ p2,
                                                       float* __restrict__ o0,
                                                       float* __restrict__ o1,
                                                       float* __restrict__ o2,
                                                       float* __restrict__ n0,
                                                       float* __restrict__ n1,
                                                       float* __restrict__ n2) {
  const int t = blockIdx.x / B;
  const int b = blockIdx.x % B;
  const float* p = (t == 0) ? p0 : (t == 1) ? p1 : p2;
  float* o = (t == 0) ? o0 : (t == 1) ? o1 : o2;
  float* nn = (t == 0) ? n0 : (t == 1) ? n1 : n2;
  const int d = threadIdx.x;
  float acc = 0.f;
  #pragma unroll
  for (int s = 0; s < SPLIT; ++s) acc += p[((size_t)(b * SPLIT + s)) * D + d];
  o[(size_t)b * D + d] = acc;
  __shared__ float red[128];
  red[d] = acc * acc;
  __syncthreads();
  for (int off = 64; off > 0; off >>= 1) {
    if (d < off) red[d] += red[d + off];
    __syncthreads();
  }
  if (d == 0) nn[b] = sqrtf(red[0]);
}

// ---------------- fused kv_insert(param) x3 + replace(col1,col2) ----------------
// wave-per-row: kv_insert rows kept in registers, wave-reduce dot/norm vs param,
// apply replace, store once. tags stored directly after insert.

__global__ void __launch_bounds__(256) kv_replace_k(const float* __restrict__ tags,
                                                    const float* __restrict__ col1,
                                                    const float* __restrict__ col2,
                                                    const float* __restrict__ cs3,
                                                    const float* __restrict__ a_tag,
                                                    const float* __restrict__ a_l,
                                                    const float* __restrict__ a_r,
                                                    const float* __restrict__ param,
                                                    const float* __restrict__ param_n,
                                                    const float* __restrict__ argv,
                                                    float* __restrict__ otags,
                                                    float* __restrict__ ocol1,
                                                    float* __restrict__ ocol2) {
  const int lane = threadIdx.x & 31;
  const int wave = threadIdx.x >> 5;
  const int row = blockIdx.x * 8 + wave;
  const int b = row >> 11;
  const size_t roff = (size_t)row * D + lane * 4;
  const size_t boff = (size_t)b * D + lane * 4;

  const f4 tg = ldg4(tags + roff);
  const f4 c1 = ldg4(col1 + roff);
  const f4 c2 = ldg4(col2 + roff);
  const f4 at4 = ldg4(a_tag + boff);
  const f4 al4 = ldg4(a_l + boff);
  const f4 ar4 = ldg4(a_r + boff);
  const f4 pm4 = ldg4(param + boff);
  const f4 ag4 = ldg4(argv + boff);

  const float a = clampA(cs3[row]);
  const f4 t0 = lerp4(tg, at4, a);
  const f4 t1 = lerp4(c1, al4, a);
  const f4 t2 = lerp4(c2, ar4, a);
  stg4(otags + roff, t0);

  const float d1 = wave_sum32(dot4(t1, pm4));
  const float s1 = wave_sum32(dot4(t1, t1));
  const float d2 = wave_sum32(dot4(t2, pm4));
  const float s2 = wave_sum32(dot4(t2, t2));
  const float pn = param_n[b];
  const float sim1 = d1 / fmaxf(pn * sqrtf(s1), EPSF);
  const float sim2 = d2 / fmaxf(pn * sqrtf(s2), EPSF);

  stg4(ocol1 + roff, lerp4(t1, ag4, sim1));
  stg4(ocol2 + roff, lerp4(t2, ag4, sim2));
}

// ---------------- final: kv_insert(at_addr) x3 + GC loop ----------------
// thread per float4 chunk; GC iterates gc_steps (read on-device) with honest zv.

__global__ void __launch_bounds__(256) finalize_k(const float* __restrict__ cs0,
                                                  const float* __restrict__ cs1,
                                                  const float* __restrict__ cs3,
                                                  const float* __restrict__ cs4,
                                                  const float* __restrict__ b_tag,
                                                  const float* __restrict__ b_l,
                                                  const float* __restrict__ b_r,
                                                  const float* __restrict__ zv,
                                                  const int* __restrict__ gc_steps,
                                                  float* __restrict__ otags,
                                                  float* __restrict__ ocol1,
                                                  float* __restrict__ ocol2) {
  const int idx = blockIdx.x * 256 + threadIdx.x; // over B*N*(D/4) chunks
  const int row = idx >> 5;
  const int d4 = (idx & 31) * 4;
  const int b = row >> 11;
  const size_t roff = (size_t)row * D + d4;
  const size_t boff = (size_t)b * D + d4;

  const float a0 = clampA(cs0[row]);
  const float a1 = clampA(cs1[row]);   // fn_addr
  const float a3 = clampA(cs3[row]);   // param_addr
  const float a4 = clampA(cs4[row]);   // body_addr
  const int gs = *gc_steps;
  const f4 z = ldg4(zv + d4);

  f4 v;
  v = lerp4(ldg4(otags + roff), ldg4(b_tag + boff), a0);
  for (int g = 0; g < gs; ++g) { v = lerp4(v, z, a1); v = lerp4(v, z, a3); v = lerp4(v, z, a4); }
  stg4(otags + roff, v);

  v = lerp4(ldg4(ocol1 + roff), ldg4(b_l + boff), a0);
  for (int g = 0; g < gs; ++g) { v = lerp4(v, z, a1); v = lerp4(v, z, a3); v = lerp4(v, z, a4); }
  stg4(ocol1 + roff, v);

  v = lerp4(ldg4(ocol2 + roff), ldg4(b_r + boff), a0);
  for (int g = 0; g < gs; ++g) { v = lerp4(v, z, a1); v = lerp4(v, z, a3); v = lerp4(v, z, a4); }
  stg4(ocol2 + roff, v);
}

// ---------------- host ----------------

extern "C" void kernel_launch(void* const* d_in, const int* in_sizes, int n_in,
                              void* d_out, int out_size, void* d_ws, size_t ws_size,
                              hipStream_t stream) {
  (void)in_sizes; (void)n_in; (void)out_size; (void)ws_size;
  const float* at_addr   = (const float*)d_in[0];
  const float* addresses = (const float*)d_in[1];
  const float* tags      = (const float*)d_in[2];
  const float* col1      = (const float*)d_in[3];
  const float* col2      = (const float*)d_in[4];
  const float* zero_vec  = (const float*)d_in[5];
  const int*   gc_steps  = (const int*)d_in[6];

  float* otags = (float*)d_out;
  float* ocol1 = otags + (size_t)B * N * D;
  float* ocol2 = ocol1 + (size_t)B * N * D;

  // workspace layout (floats)
  float* ws = (float*)d_ws;
  size_t o = 0;
  float* at_norm   = ws + o; o += B;
  float* fn_n      = ws + o; o += B;
  float* arg_n     = ws + o; o += B;
  float* param_n   = ws + o; o += B;
  float* body_n    = ws + o; o += B;
  float* dn0       = ws + o; o += B;   // dummy norm sinks
  float* dn1       = ws + o; o += B;
  float* dn2       = ws + o; o += B;
  float* addr_norm = ws + o; o += (size_t)B * N;
  float* cs0       = ws + o; o += (size_t)B * N;
  float* cs1       = ws + o; o += (size_t)B * N;
  float* cs2       = ws + o; o += (size_t)B * N;
  float* cs3       = ws + o; o += (size_t)B * N;
  float* cs4       = ws + o; o += (size_t)B * N;
  float* fn_addr   = ws + o; o += (size_t)B * D;
  float* arg_addr  = ws + o; o += (size_t)B * D;
  float* param_a   = ws + o; o += (size_t)B * D;
  float* body_a    = ws + o; o += (size_t)B * D;
  float* a_tag     = ws + o; o += (size_t)B * D;
  float* a_l       = ws + o; o += (size_t)B * D;
  float* a_r       = ws + o; o += (size_t)B * D;
  float* b_tag     = ws + o; o += (size_t)B * D;
  float* b_l       = ws + o; o += (size_t)B * D;
  float* b_r       = ws + o; o += (size_t)B * D;
  float* p0        = ws + o; o += (size_t)B * SPLIT * D;
  float* p1        = ws + o; o += (size_t)B * SPLIT * D;
  float* p2        = ws + o; o += (size_t)B * SPLIT * D;

  const int rowBlocks = (B * N) / 8;       // wave-per-row kernels
  const int wmmaBlocks = B * (N / 16);     // 4096
  const int selBlocks = B * SPLIT;

  // 1. ||at_addr||
  vec_norm_k<<<B, 128, 0, stream>>>(at_addr, at_norm);
  // 2. row norms + cs0
  rownorm_cs0_k<<<rowBlocks, 256, 0, stream>>>(addresses, at_addr, at_norm,
                                               addr_norm, cs0);
  // 3. fn_addr, arg_addr = select(cs0; col1, col2)
  select_split_k<<<selBlocks, 128, 0, stream>>>(cs0, col1, col2, nullptr, p0, p1, nullptr);
  select_reduce_k<<<2 * B, 128, 0, stream>>>(p0, p1, nullptr, fn_addr, arg_addr, nullptr,
                                             fn_n, arg_n, nullptr);
  // 4. cs1 (fn), cs2 (arg) via WMMA
  wmma_dots2_k<<<wmmaBlocks, 32, 0, stream>>>(addresses, addr_norm, fn_addr, arg_addr,
                                              fn_n, arg_n, cs1, cs2);
  // 5. param_addr, body_addr = select(cs1; col1, col2)   (fn_tag is dead)
  select_split_k<<<selBlocks, 128, 0, stream>>>(cs1, col1, col2, nullptr, p0, p1, nullptr);
  select_reduce_k<<<2 * B, 128, 0, stream>>>(p0, p1, nullptr, param_a, body_a, nullptr,
                                             param_n, body_n, nullptr);
  // 6. a_tag, a_l, a_r = select(cs2; tags, col1, col2)
  select_split_k<<<selBlocks, 128, 0, stream>>>(cs2, tags, col1, col2, p0, p1, p2);
  select_reduce_k<<<3 * B, 128, 0, stream>>>(p0, p1, p2, a_tag, a_l, a_r, dn0, dn1, dn2);
  // 7. cs3 (param), cs4 (body) via WMMA
  wmma_dots2_k<<<wmmaBlocks, 32, 0, stream>>>(addresses, addr_norm, param_a, body_a,
                                              param_n, body_n, cs3, cs4);
  // 8. fused kv_insert(param) + replace -> d_out regions
  kv_replace_k<<<rowBlocks, 256, 0, stream>>>(tags, col1, col2, cs3, a_tag, a_l, a_r,
                                              param_a, param_n, arg_addr,
                                              otags, ocol1, ocol2);
  // 9. b_tag, b_l, b_r = select(cs4; updated tensors)
  select_split_k<<<selBlocks, 128, 0, stream>>>(cs4, otags, ocol1, ocol2, p0, p1, p2);
  select_reduce_k<<<3 * B, 128, 0, stream>>>(p0, p1, p2, b_tag, b_l, b_r, dn0, dn1, dn2);
  // 10. fused kv_insert(at_addr) + GC loop (in place on d_out)
  finalize_k<<<(B * N * (D / 4)) / 256, 256, 0, stream>>>(cs0, cs1, cs3, cs4,
                                                          b_tag, b_l, b_r,
                                                          zero_vec, gc_steps,
                                                          otags, ocol1, ocol2);
}